// ScalarInteractionModule_61735859912707
// MI455X (gfx1250) — compile-verified
//
#include <hip/hip_runtime.h>
#include <hip/hip_bf16.h>
#include <stdint.h>

// ---------------------------------------------------------------------------
// ScalarInteractionModule on MI455X (gfx1250).
// latents = uc*c*(1+c+c^2) * upd  (recycle loop collapses; see round-0 notes).
// GEMMs via v_wmma_f32_16x16x4_f32, 32-edge M-tiles per wave (2 A-frags share
// each B-frag). Index gathers use global_load_async_to_lds_b128 (ASYNCcnt).
// ---------------------------------------------------------------------------

#define NN      20000      // N_NODES
#define NE      640000     // N_EDGES
#define NB      8          // NUM_BASIS
#define AD      64         // ATTR_DIM
#define LAT     64         // LATENT
#define D1      136        // 2*ATTR_DIM + NUM_BASIS
#define LN_EPS  1e-5f
#define RMAXF   6.0f
#define PI_F    3.14159265358979f

typedef __attribute__((ext_vector_type(2))) float v2f;
typedef __attribute__((ext_vector_type(8))) float v8f;

__device__ __forceinline__ v8f wmma_f32_k4(v2f a, v2f b, v8f c) {
  return __builtin_amdgcn_wmma_f32_16x16x4_f32(
      false, a, false, b, (short)0, c, false, false);
}

__device__ __forceinline__ float sigmoidf_(float x) {
  return 1.0f / (1.0f + __expf(-x));
}

// Low 32 bits of a generic pointer into __shared__ = LDS byte offset
// (flat->LDS mapping truncates the aperture bits, ISA 10.2).
__device__ __forceinline__ unsigned lds_off(const void* p) {
  return (unsigned)(uintptr_t)p;
}

// Async 16B global->LDS copy, tracked by ASYNCcnt.
__device__ __forceinline__ void async_ld_b128(unsigned ldsdst, const void* src) {
  asm volatile("global_load_async_to_lds_b128 %0, %1, off"
               :: "v"(ldsdst), "v"(src)
               : "memory");
}
__device__ __forceinline__ void async_wait0() {
  asm volatile("s_wait_asynccnt 0" ::: "memory");
}

// ---------------------------------------------------------------------------
// 1) Radial basis
// ---------------------------------------------------------------------------
__global__ void rbf_kernel(const float* __restrict__ elen, float* __restrict__ rbf) {
  int e = blockIdx.x * 256 + threadIdx.x;
  if (e >= NE) return;
  float r  = elen[e];
  float x  = r * (1.0f / RMAXF);
  float x2 = x * x;
  float x6 = x2 * x2 * x2;
  float th = tanhf(1.0f - x6);
  float cut = th * th * th;
  float pref = sqrtf(2.0f / RMAXF) * cut / r;
  float base = PI_F * x;
#pragma unroll
  for (int n = 1; n <= NB; ++n)
    rbf[e * NB + (n - 1)] = pref * __sinf(base * (float)n);
}

// ---------------------------------------------------------------------------
// 2) ef = LN1(a_c || a_n || rbf) @ W1 + b1  (WMMA, K=136, M-tile=32/wave)
//    score = LN2(ef || a_c) @ W2 + b2       (VALU)
// Block = 64 threads = 2 waves = 64 edges; grid = 10000.
// ---------------------------------------------------------------------------
__global__ __launch_bounds__(64) void ef_score_kernel(
    const long long* __restrict__ eidx, const float* __restrict__ na,
    const float* __restrict__ rbf,
    const float* __restrict__ ln1_g, const float* __restrict__ ln1_b,
    const float* __restrict__ W1, const float* __restrict__ b1,
    const float* __restrict__ ln2_g, const float* __restrict__ ln2_b,
    const float* __restrict__ W2, const float* __restrict__ b2,
    float* __restrict__ ef_out, float* __restrict__ scores) {
  __shared__ float xs[2][32 * D1];      // per-wave: edge_in, then ef[32][64]
  __shared__ float stat[2][2][32];
  const int wave = threadIdx.x >> 5;
  const int lane = threadIdx.x & 31;
  const int e0 = blockIdx.x * 64 + wave * 32;
  float* x = xs[wave];

  // Phase 1: async-gather edge_in in 16B chunks straight into LDS.
  {
    const int part = lane >> 4;    // 0 = a_c, 1 = a_n
    const int q    = lane & 15;    // which 16B chunk of the 64-float row
#pragma unroll 4
    for (int e = 0; e < 32; ++e) {
      int ge = e0 + e;
      long long node = eidx[part * NE + ge];
      const float* src = na + (size_t)node * AD + q * 4;
      async_ld_b128(lds_off(x + e * D1 + part * AD + q * 4), src);
    }
#pragma unroll
    for (int i = 0; i < 2; ++i) {  // rbf tail: 2 chunks per edge
      int c = lane + 32 * i;
      int e = c >> 1, half = c & 1;
      const float* src = rbf + (size_t)(e0 + e) * NB + half * 4;
      async_ld_b128(lds_off(x + e * D1 + 2 * AD + half * 4), src);
    }
    async_wait0();
  }
  __syncthreads();

  // Phase 2: LN1 stats (2 lanes/edge, two 16-edge halves).
#pragma unroll
  for (int h = 0; h < 2; ++h) {
    int e = (lane >> 1) + h * 16, p = lane & 1;
    float s = 0.f, ss = 0.f;
    for (int d = p * 68; d < p * 68 + 68; ++d) {
      float v = x[e * D1 + d]; s += v; ss += v * v;
    }
    s  += __shfl_xor(s, 1, 32);
    ss += __shfl_xor(ss, 1, 32);
    float m = s * (1.0f / (float)D1);
    float rstd = rsqrtf(ss * (1.0f / (float)D1) - m * m + LN_EPS);
    if (p == 0) { stat[wave][0][e] = m; stat[wave][1][e] = rstd; }
  }
  __syncthreads();

  // Phase 3: normalize in place.
  for (int idx = lane; idx < 32 * D1; idx += 32) {
    int e = idx / D1, d = idx - e * D1;
    x[idx] = (x[idx] - stat[wave][0][e]) * stat[wave][1][e] * ln1_g[d] + ln1_b[d];
  }
  __syncthreads();

  // Phase 4: WMMA GEMM, 32x64 tile, K=136; two A-frags share each B-frag.
  const int khalf = lane >> 4;
  const int nlo   = lane & 15;
  const int row   = lane & 15;
  v8f acc[2][4];
#pragma unroll
  for (int nt = 0; nt < 4; ++nt) {
    float bv = b1[nt * 16 + nlo];
    acc[0][nt] = (v8f){bv, bv, bv, bv, bv, bv, bv, bv};
    acc[1][nt] = acc[0][nt];
  }
  for (int k0 = 0; k0 < D1; k0 += 4) {
    v2f a0, a1;
    a0.x = x[row * D1 + k0 + 2 * khalf];
    a0.y = x[row * D1 + k0 + 2 * khalf + 1];
    a1.x = x[(row + 16) * D1 + k0 + 2 * khalf];
    a1.y = x[(row + 16) * D1 + k0 + 2 * khalf + 1];
#pragma unroll
    for (int nt = 0; nt < 4; ++nt) {
      int n = nt * 16 + nlo;
      v2f b;
      b.x = W1[(k0 + 2 * khalf) * LAT + n];
      b.y = W1[(k0 + 2 * khalf + 1) * LAT + n];
      acc[0][nt] = wmma_f32_k4(a0, b, acc[0][nt]);
      acc[1][nt] = wmma_f32_k4(a1, b, acc[1][nt]);
    }
  }
  __syncthreads();

  // Phase 5: unpack C layout -> LDS ef[32][64] + global ef.
#pragma unroll
  for (int mh = 0; mh < 2; ++mh) {
#pragma unroll
    for (int nt = 0; nt < 4; ++nt) {
      int n = nt * 16 + nlo;
#pragma unroll
      for (int i = 0; i < 8; ++i) {
        int r = mh * 16 + i + 8 * khalf;
        float v = acc[mh][nt][i];
        x[r * LAT + n] = v;
        ef_out[(size_t)(e0 + r) * LAT + n] = v;
      }
    }
  }
  __syncthreads();

  // Phase 6: attention score (2 lanes/edge, two halves).
#pragma unroll
  for (int h = 0; h < 2; ++h) {
    int e = (lane >> 1) + h * 16, p = lane & 1;
    int ge = e0 + e;
    int c = (int)eidx[ge];
    float s = 0.f, ss = 0.f;
    for (int d = 0; d < AD; ++d) {
      float v = p ? na[c * AD + d] : x[e * LAT + d];
      s += v; ss += v * v;
    }
    s  += __shfl_xor(s, 1, 32);
    ss += __shfl_xor(ss, 1, 32);
    float m = s * (1.0f / 128.0f);
    float rstd = rsqrtf(ss * (1.0f / 128.0f) - m * m + LN_EPS);
    float dot = 0.f;
    for (int d = 0; d < AD; ++d) {
      int gd = p * AD + d;
      float v = p ? na[c * AD + d] : x[e * LAT + d];
      dot += ((v - m) * rstd * ln2_g[gd] + ln2_b[gd]) * W2[gd];
    }
    dot += __shfl_xor(dot, 1, 32);
    if (p == 0) scores[ge] = dot + b2[0];
  }
}

// ---------------------------------------------------------------------------
// 3) Global softmax denominators.
// ---------------------------------------------------------------------------
__global__ void reduce_max_part(const float* __restrict__ s, float* __restrict__ part) {
  __shared__ float sm[256];
  float m = -3.4e38f;
  for (int i = blockIdx.x * 256 + threadIdx.x; i < NE; i += 256 * 256)
    m = fmaxf(m, s[i]);
  sm[threadIdx.x] = m; __syncthreads();
  for (int k = 128; k > 0; k >>= 1) {
    if (threadIdx.x < k) sm[threadIdx.x] = fmaxf(sm[threadIdx.x], sm[threadIdx.x + k]);
    __syncthreads();
  }
  if (threadIdx.x == 0) part[blockIdx.x] = sm[0];
}
__global__ void reduce_max_final(const float* __restrict__ part, float* __restrict__ stats) {
  __shared__ float sm[256];
  sm[threadIdx.x] = part[threadIdx.x]; __syncthreads();
  for (int k = 128; k > 0; k >>= 1) {
    if (threadIdx.x < k) sm[threadIdx.x] = fmaxf(sm[threadIdx.x], sm[threadIdx.x + k]);
    __syncthreads();
  }
  if (threadIdx.x == 0) stats[0] = sm[0];
}
__global__ void reduce_sum_part(const float* __restrict__ s, const float* __restrict__ stats,
                                float* __restrict__ part) {
  __shared__ float sm[256];
  float gmax = stats[0];
  float acc = 0.f;
  for (int i = blockIdx.x * 256 + threadIdx.x; i < NE; i += 256 * 256)
    acc += __expf(s[i] - gmax);
  sm[threadIdx.x] = acc; __syncthreads();
  for (int k = 128; k > 0; k >>= 1) {
    if (threadIdx.x < k) sm[threadIdx.x] += sm[threadIdx.x + k];
    __syncthreads();
  }
  if (threadIdx.x == 0) part[blockIdx.x] = sm[0];
}
__global__ void reduce_sum_final(const float* __restrict__ part, float* __restrict__ stats) {
  __shared__ float sm[256];
  sm[threadIdx.x] = part[threadIdx.x]; __syncthreads();
  for (int k = 128; k > 0; k >>= 1) {
    if (threadIdx.x < k) sm[threadIdx.x] += sm[threadIdx.x + k];
    __syncthreads();
  }
  if (threadIdx.x == 0) stats[1] = 1.0f / sm[0];
}

// ---------------------------------------------------------------------------
// 4) pooled[ec] += attn * ef  (f32 global atomics; pooled is L2-resident)
// ---------------------------------------------------------------------------
__global__ void zero_pooled(float* __restrict__ pooled) {
  int i = blockIdx.x * 256 + threadIdx.x;
  if (i < NN * LAT) pooled[i] = 0.f;
}
__global__ void pooled_scatter(const long long* __restrict__ eidx,
                               const float* __restrict__ scores,
                               const float* __restrict__ stats,
                               const float* __restrict__ ef,
                               float* __restrict__ pooled) {
  int tid = blockIdx.x * 256 + threadIdx.x;  // NE*64 threads exactly
  int e = tid >> 6, d = tid & 63;
  float attn = __expf(scores[e] - stats[0]) * stats[1];
  int c = (int)eidx[e];
  atomicAdd(&pooled[c * LAT + d], attn * ef[(size_t)e * LAT + d]);
}

// ---------------------------------------------------------------------------
// 5) gate = sigmoid((silu(a)*g) @ film_W2); upd = LN3(ef || pooled[ec]) @ W3 + b3
//    latents = factor * upd * gate   (two WMMA GEMMs, K=128, M-tile=32/wave)
// ---------------------------------------------------------------------------
__global__ __launch_bounds__(64) void gate_upd_kernel(
    const long long* __restrict__ eidx, const float* __restrict__ rbf,
    const float* __restrict__ film_W1, const float* __restrict__ film_W2,
    const float* __restrict__ ef, const float* __restrict__ pooled,
    const float* __restrict__ ln3_g, const float* __restrict__ ln3_b,
    const float* __restrict__ W3, const float* __restrict__ b3,
    const float* __restrict__ resnet_p, float* __restrict__ out) {
  __shared__ float ts[2][32 * 128];    // per-wave: film product, then upd_in
  __shared__ float stat[2][2][32];
  const int wave = threadIdx.x >> 5;
  const int lane = threadIdx.x & 31;
  const int e0 = blockIdx.x * 64 + wave * 32;
  float* t = ts[wave];

  // Phase 1: h = rbf @ film_W1 (K=8, VALU); t = silu(a)*g.
#pragma unroll
  for (int h = 0; h < 2; ++h) {
    int e = (lane >> 1) + h * 16, p = lane & 1;
    int ge = e0 + e;
    float rb[NB];
#pragma unroll
    for (int k = 0; k < NB; ++k) rb[k] = rbf[ge * NB + k];
    for (int j = 0; j < 64; ++j) {
      int jj = p * 64 + j;
      float av = 0.f, gv = 0.f;
#pragma unroll
      for (int k = 0; k < NB; ++k) {
        av += rb[k] * film_W1[k * 256 + jj];
        gv += rb[k] * film_W1[k * 256 + 128 + jj];
      }
      t[e * 128 + jj] = (av * sigmoidf_(av)) * gv;
    }
  }
  __syncthreads();

  // Phase 2: gate GEMM (K=128) vs film_W2.
  const int khalf = lane >> 4;
  const int nlo   = lane & 15;
  const int row   = lane & 15;
  v8f accg[2][4];
#pragma unroll
  for (int nt = 0; nt < 4; ++nt) {
    accg[0][nt] = (v8f){0, 0, 0, 0, 0, 0, 0, 0};
    accg[1][nt] = accg[0][nt];
  }
  for (int k0 = 0; k0 < 128; k0 += 4) {
    v2f a0, a1;
    a0.x = t[row * 128 + k0 + 2 * khalf];
    a0.y = t[row * 128 + k0 + 2 * khalf + 1];
    a1.x = t[(row + 16) * 128 + k0 + 2 * khalf];
    a1.y = t[(row + 16) * 128 + k0 + 2 * khalf + 1];
#pragma unroll
    for (int nt = 0; nt < 4; ++nt) {
      int n = nt * 16 + nlo;
      v2f b;
      b.x = film_W2[(k0 + 2 * khalf) * LAT + n];
      b.y = film_W2[(k0 + 2 * khalf + 1) * LAT + n];
      accg[0][nt] = wmma_f32_k4(a0, b, accg[0][nt]);
      accg[1][nt] = wmma_f32_k4(a1, b, accg[1][nt]);
    }
  }
  __syncthreads();

  // Phase 3: async-gather upd_in = ef || pooled[ec] into the same LDS region.
  {
    const int part = lane >> 4;
    const int q    = lane & 15;
#pragma unroll 4
    for (int e = 0; e < 32; ++e) {
      int ge = e0 + e;
      const float* src;
      if (part == 0) {
        src = ef + (size_t)ge * LAT + q * 4;
      } else {
        long long c = eidx[ge];
        src = pooled + (size_t)c * LAT + q * 4;
      }
      async_ld_b128(lds_off(t + e * 128 + part * AD + q * 4), src);
    }
    async_wait0();
  }
  __syncthreads();

  // Phase 4: LN3 stats + normalize.
#pragma unroll
  for (int h = 0; h < 2; ++h) {
    int e = (lane >> 1) + h * 16, p = lane & 1;
    float s = 0.f, ss = 0.f;
    for (int d = p * 64; d < p * 64 + 64; ++d) {
      float v = t[e * 128 + d]; s += v; ss += v * v;
    }
    s  += __shfl_xor(s, 1, 32);
    ss += __shfl_xor(ss, 1, 32);
    float m = s * (1.0f / 128.0f);
    float rstd = rsqrtf(ss * (1.0f / 128.0f) - m * m + LN_EPS);
    if (p == 0) { stat[wave][0][e] = m; stat[wave][1][e] = rstd; }
  }
  __syncthreads();
  for (int idx = lane; idx < 32 * 128; idx += 32) {
    int e = idx >> 7, d = idx & 127;
    t[idx] = (t[idx] - stat[wave][0][e]) * stat[wave][1][e] * ln3_g[d] + ln3_b[d];
  }
  __syncthreads();

  // Phase 5: upd GEMM (K=128) vs W3, C init = b3.
  v8f accu[2][4];
#pragma unroll
  for (int nt = 0; nt < 4; ++nt) {
    float bv = b3[nt * 16 + nlo];
    accu[0][nt] = (v8f){bv, bv, bv, bv, bv, bv, bv, bv};
    accu[1][nt] = accu[0][nt];
  }
  for (int k0 = 0; k0 < 128; k0 += 4) {
    v2f a0, a1;
    a0.x = t[row * 128 + k0 + 2 * khalf];
    a0.y = t[row * 128 + k0 + 2 * khalf + 1];
    a1.x = t[(row + 16) * 128 + k0 + 2 * khalf];
    a1.y = t[(row + 16) * 128 + k0 + 2 * khalf + 1];
#pragma unroll
    for (int nt = 0; nt < 4; ++nt) {
      int n = nt * 16 + nlo;
      v2f b;
      b.x = W3[(k0 + 2 * khalf) * LAT + n];
      b.y = W3[(k0 + 2 * khalf + 1) * LAT + n];
      accu[0][nt] = wmma_f32_k4(a0, b, accu[0][nt]);
      accu[1][nt] = wmma_f32_k4(a1, b, accu[1][nt]);
    }
  }

  // Phase 6: collapsed-recycle scale + gate, write latents.
  float uc = sigmoidf_(resnet_p[0]);
  float c  = rsqrtf(uc * uc + 1.0f);
  float factor = uc * c * (1.0f + c + c * c);
#pragma unroll
  for (int mh = 0; mh < 2; ++mh) {
#pragma unroll
    for (int nt = 0; nt < 4; ++nt) {
      int n = nt * 16 + nlo;
#pragma unroll
      for (int i = 0; i < 8; ++i) {
        int r = mh * 16 + i + 8 * khalf;
        float gate = sigmoidf_(accg[mh][nt][i]);
        out[(size_t)(e0 + r) * LAT + n] = factor * accu[mh][nt][i] * gate;
      }
    }
  }
}

// ---------------------------------------------------------------------------
// Host-side launch. Workspace (floats):
//   ef[NE*64] | rbf[NE*8] | scores[NE] | pooled[NN*64] | partials[512] | stats[2]
// ---------------------------------------------------------------------------
extern "C" void kernel_launch(void* const* d_in, const int* in_sizes, int n_in,
                              void* d_out, int out_size, void* d_ws, size_t ws_size,
                              hipStream_t stream) {
  const long long* eidx   = (const long long*)d_in[0];
  const float* elen       = (const float*)d_in[1];
  const float* na         = (const float*)d_in[2];
  const float* ln1_g      = (const float*)d_in[3];
  const float* ln1_b      = (const float*)d_in[4];
  const float* W1         = (const float*)d_in[5];
  const float* b1         = (const float*)d_in[6];
  const float* ln2_g      = (const float*)d_in[7];
  const float* ln2_b      = (const float*)d_in[8];
  const float* W2         = (const float*)d_in[9];
  const float* b2         = (const float*)d_in[10];
  const float* ln3_g      = (const float*)d_in[11];
  const float* ln3_b      = (const float*)d_in[12];
  const float* W3         = (const float*)d_in[13];
  const float* b3         = (const float*)d_in[14];
  const float* film_W1    = (const float*)d_in[15];
  const float* film_W2    = (const float*)d_in[16];
  const float* resnet_p   = (const float*)d_in[17];
  float* out = (float*)d_out;

  float* ws = (float*)d_ws;
  float* ef      = ws;
  float* rbf     = ef + (size_t)NE * LAT;
  float* scores  = rbf + (size_t)NE * NB;
  float* pooled  = scores + (size_t)NE;
  float* partmax = pooled + (size_t)NN * LAT;
  float* partsum = partmax + 256;
  float* stats   = partsum + 256;

  rbf_kernel<<<(NE + 255) / 256, 256, 0, stream>>>(elen, rbf);

  ef_score_kernel<<<NE / 64, 64, 0, stream>>>(
      eidx, na, rbf, ln1_g, ln1_b, W1, b1, ln2_g, ln2_b, W2, b2, ef, scores);

  reduce_max_part<<<256, 256, 0, stream>>>(scores, partmax);
  reduce_max_final<<<1, 256, 0, stream>>>(partmax, stats);
  reduce_sum_part<<<256, 256, 0, stream>>>(scores, stats, partsum);
  reduce_sum_final<<<1, 256, 0, stream>>>(partsum, stats);

  zero_pooled<<<(NN * LAT + 255) / 256, 256, 0, stream>>>(pooled);
  pooled_scatter<<<(NE * LAT) / 256, 256, 0, stream>>>(eidx, scores, stats, ef, pooled);

  gate_upd_kernel<<<NE / 64, 64, 0, stream>>>(
      eidx, rbf, film_W1, film_W2, ef, pooled, ln3_g, ln3_b, W3, b3, resnet_p, out);
}